// GATLayer_66108136620616
// MI455X (gfx1250) — compile-verified
//
#include <hip/hip_runtime.h>
#include <hip/hip_bf16.h>
#include <math.h>

#define B_    4
#define N_    1024
#define IN_   128
#define OUT_  128
#define E_    16
#define EP_   20          // padded edge row (floats) to spread LDS banks
#define H_    4
#define HD_   32
#define BN_   (B_ * N_)
#define LN_EPS 1e-5f

typedef float v8f __attribute__((ext_vector_type(8)));
typedef float v2f __attribute__((ext_vector_type(2)));

__device__ __forceinline__ v8f wmma_f32_4(v2f a, v2f b, v8f c) {
  // D = A(16x4 f32) * B(4x16 f32) + C(16x16 f32)
  return __builtin_amdgcn_wmma_f32_16x16x4_f32(false, a, false, b, (short)0, c,
                                               false, false);
}

// ---------------------------------------------------------------------------
// Kernel 1: Wh = h @ W   (M=BN=4096, K=128, N=128), fp32 WMMA 16x16x4.
// One block = one 16-row tile; 8 waves, each owning a 16-col tile.
// ---------------------------------------------------------------------------
__global__ __launch_bounds__(256) void wh_kernel(const float* __restrict__ h,
                                                 const float* __restrict__ W,
                                                 float* __restrict__ Wh) {
  const int i0   = blockIdx.x * 16;
  const int wave = threadIdx.x >> 5;
  const int lane = threadIdx.x & 31;
  const int col  = lane & 15;
  const int koff = (lane >> 4) * 2;
  const int d0   = wave * 16;

  v8f acc = {};
  for (int kb = 0; kb < IN_; kb += 4) {
    v2f a, b;
    const float* hp = h + (size_t)(i0 + col) * IN_ + kb + koff;
    a.x = hp[0];
    a.y = hp[1];
    const float* wp = W + (size_t)(kb + koff) * OUT_ + d0 + col;
    b.x = wp[0];
    b.y = wp[OUT_];
    acc = wmma_f32_4(a, b, acc);
  }
#pragma unroll
  for (int v = 0; v < 8; ++v) {
    const int row = v + ((lane >> 4) << 3);
    Wh[(size_t)(i0 + row) * OUT_ + d0 + col] = acc[v];
  }
}

// ---------------------------------------------------------------------------
// Kernel 2: per-node attention projections (tiny VALU kernel).
// ---------------------------------------------------------------------------
__global__ __launch_bounds__(256) void score_kernel(
    const float* __restrict__ Wh, const float* __restrict__ a_src,
    const float* __restrict__ a_tgt, float* __restrict__ s_src,
    float* __restrict__ s_tgt) {
  const int idx = blockIdx.x * blockDim.x + threadIdx.x;  // n*H + h
  if (idx >= BN_ * H_) return;
  const int n  = idx >> 2;
  const int hh = idx & 3;
  const float* wp = Wh + (size_t)n * OUT_ + hh * HD_;
  float ss = 0.f, st = 0.f;
#pragma unroll
  for (int d = 0; d < HD_; ++d) {
    const float w = wp[d];
    ss += w * a_src[hh * HD_ + d];
    st += w * a_tgt[hh * HD_ + d];
  }
  s_src[idx] = ss;
  s_tgt[idx] = st;
}

// ---------------------------------------------------------------------------
// Kernel 3: fused flash-style GAT, double-buffered async edge staging.
// One block = one 16-row i-tile; 4 waves = 4 heads. edge_feats read ONCE
// from HBM and ONCE into LDS (shared by all heads), streamed with
// global_load_async_to_lds_b128 + s_wait_asynccnt while WMMAs consume
// the previous tile.
// ---------------------------------------------------------------------------
__global__ __launch_bounds__(128) void gat_kernel(
    const float* __restrict__ hin, const float* __restrict__ edge,
    const float* __restrict__ Wh, const float* __restrict__ s_src,
    const float* __restrict__ s_tgt, const float* __restrict__ We,
    const float* __restrict__ gamma, const float* __restrict__ beta,
    float* __restrict__ out) {
  __shared__ float Ebuf[2][16][16][EP_];  // double-buffered edge tile (40 KB)
  __shared__ float Pbuf[H_][16][16];      // per-wave exp(score) tile
  __shared__ float Obuf[16][OUT_];        // residual-added pre-LN output
  __shared__ float lbuf[H_][16];          // per-row softmax denominators
  __shared__ float mu_s[16], rs_s[16];

  const int tid  = threadIdx.x;
  const int i0   = blockIdx.x * 16;       // row tile in flattened (B*N)
  const int b    = i0 >> 10;              // N = 1024
  const int hh   = tid >> 5;              // wave id == head id
  const int lane = tid & 31;
  const int col  = lane & 15;
  const int koff = (lane >> 4) * 2;

  // Per-thread async-copy descriptor: thread owns (r = 2q + rhalf, c, e..e+3)
  const int rhalf = tid >> 6;             // 0/1
  const int ec    = (tid >> 2) & 15;      // tile column
  const int ee    = (tid & 3) * 4;        // float offset within edge row
  const unsigned ldsE0 =
      (unsigned)(size_t)(void*)&Ebuf[0][0][0][0];  // LDS offset (low 32 bits)
  const unsigned bufBytes = 16 * 16 * EP_ * 4;     // one buffer

  // We column for this head, in registers.
  float wecol[E_];
#pragma unroll
  for (int e = 0; e < E_; ++e) wecol[e] = We[e * H_ + hh];

  v8f acc0 = {};
  v8f acc1 = {};
  float lsum = 0.f;

  // --- prologue: async-stage tile j0=0 into buffer 0 ---
#pragma unroll
  for (int q = 0; q < 8; ++q) {
    const int r = 2 * q + rhalf;
    const float* g = edge + ((size_t)(i0 + r) * N_ + ec) * E_ + ee;
    const unsigned l = ldsE0 + (unsigned)(((r * 16 + ec) * EP_ + ee) * 4);
    asm volatile("global_load_async_to_lds_b128 %0, %1, off"
                 :: "v"(l), "v"(g) : "memory");
  }

  for (int jt = 0; jt < N_ / 16; ++jt) {
    const int j0  = jt * 16;
    const int buf = jt & 1;

    if (jt + 1 < N_ / 16) {
      // issue next tile into the other buffer
      const unsigned lbase = ldsE0 + (unsigned)(buf ^ 1) * bufBytes;
#pragma unroll
      for (int q = 0; q < 8; ++q) {
        const int r = 2 * q + rhalf;
        const float* g =
            edge + ((size_t)(i0 + r) * N_ + (j0 + 16 + ec)) * E_ + ee;
        const unsigned l = lbase + (unsigned)(((r * 16 + ec) * EP_ + ee) * 4);
        asm volatile("global_load_async_to_lds_b128 %0, %1, off"
                     :: "v"(l), "v"(g) : "memory");
      }
      // async loads complete in order: <=8 outstanding => current tile done
      asm volatile("s_wait_asynccnt 0x8" ::: "memory");
    } else {
      asm volatile("s_wait_asynccnt 0x0" ::: "memory");
    }
    __syncthreads();  // tile jt visible to all 4 waves

    // --- phase A: scores from LDS edge tile (each wave: its head) ---
#pragma unroll
    for (int sub = 0; sub < 8; ++sub) {
      const int p = sub * 32 + lane;  // 0..255 -> (r,c)
      const int r = p >> 4;
      const int c = p & 15;
      const float4* ev = (const float4*)&Ebuf[buf][r][c][0];
      const float4 e0 = ev[0], e1 = ev[1], e2 = ev[2], e3 = ev[3];
      float s = e0.x * wecol[0] + e0.y * wecol[1] + e0.z * wecol[2] +
                e0.w * wecol[3] + e1.x * wecol[4] + e1.y * wecol[5] +
                e1.z * wecol[6] + e1.w * wecol[7] + e2.x * wecol[8] +
                e2.y * wecol[9] + e2.z * wecol[10] + e2.w * wecol[11] +
                e3.x * wecol[12] + e3.y * wecol[13] + e3.z * wecol[14] +
                e3.w * wecol[15];
      s += s_src[(size_t)(i0 + r) * H_ + hh] +
           s_tgt[((size_t)b * N_ + j0 + c) * H_ + hh];
      s = fmaxf(s, 0.2f * s);              // leaky_relu(0.2)
      Pbuf[hh][r][c] = __expf(s);
    }
    // --- phase B: row sums (LDS in-order within wave, wave32 lockstep) ---
    if (lane < 16) {
      float rs = 0.f;
#pragma unroll
      for (int c2 = 0; c2 < 16; ++c2) rs += Pbuf[hh][lane][c2];
      lsum += rs;
    }
    // --- phase C: acc += P(16x16) @ Wh(16x32), 8 fp32 WMMAs ---
#pragma unroll
    for (int kc = 0; kc < 4; ++kc) {
      v2f a;
      a.x = Pbuf[hh][col][kc * 4 + koff];
      a.y = Pbuf[hh][col][kc * 4 + koff + 1];
      const size_t gj = (size_t)b * N_ + j0 + kc * 4 + koff;
      const float* wp = Wh + gj * OUT_ + hh * HD_;
      v2f b0, b1;
      b0.x = wp[col];
      b0.y = wp[OUT_ + col];
      b1.x = wp[16 + col];
      b1.y = wp[OUT_ + 16 + col];
      acc0 = wmma_f32_4(a, b0, acc0);
      acc1 = wmma_f32_4(a, b1, acc1);
    }
    __syncthreads();  // all waves done with Ebuf[buf] before it is re-filled
  }

  if (lane < 16) lbuf[hh][lane] = lsum;
  __syncthreads();

  // --- epilogue: normalize, residual, stage into Obuf ---
#pragma unroll
  for (int v = 0; v < 8; ++v) {
    const int row  = v + ((lane >> 4) << 3);
    const float rc = 1.0f / lbuf[hh][row];
    const float* hp = hin + (size_t)(i0 + row) * IN_ + hh * HD_;
    Obuf[row][hh * HD_ + col]      = acc0[v] * rc + hp[col];
    Obuf[row][hh * HD_ + 16 + col] = acc1[v] * rc + hp[16 + col];
  }
  __syncthreads();

  // --- LayerNorm stats: 16 threads, one row each ---
  if (tid < 16) {
    const int row = tid;
    float m = 0.f;
    for (int c2 = 0; c2 < OUT_; ++c2) m += Obuf[row][c2];
    m *= (1.0f / OUT_);
    float var = 0.f;
    for (int c2 = 0; c2 < OUT_; ++c2) {
      const float d = Obuf[row][c2] - m;
      var += d * d;
    }
    var *= (1.0f / OUT_);
    mu_s[row] = m;
    rs_s[row] = rsqrtf(var + LN_EPS);
  }
  __syncthreads();

  // --- write out: each of 128 threads handles 16 contiguous elements ---
  {
    const int row = tid >> 3;
    const int c0  = (tid & 7) * 16;
    const float m = mu_s[row];
    const float r = rs_s[row];
    float* op = out + (size_t)(i0 + row) * OUT_;
#pragma unroll
    for (int c2 = 0; c2 < 16; ++c2) {
      const int c = c0 + c2;
      op[c] = (Obuf[row][c] - m) * r * gamma[c] + beta[c];
    }
  }
}

// ---------------------------------------------------------------------------
extern "C" void kernel_launch(void* const* d_in, const int* in_sizes, int n_in,
                              void* d_out, int out_size, void* d_ws,
                              size_t ws_size, hipStream_t stream) {
  const float* h    = (const float*)d_in[0];
  const float* ef   = (const float*)d_in[1];
  const float* W    = (const float*)d_in[2];
  const float* a_s  = (const float*)d_in[3];
  const float* a_t  = (const float*)d_in[4];
  const float* We   = (const float*)d_in[5];
  const float* gm   = (const float*)d_in[6];
  const float* bt   = (const float*)d_in[7];
  float* out = (float*)d_out;

  float* Wh   = (float*)d_ws;                    // BN*128 floats (2 MB)
  float* ssrc = Wh + (size_t)BN_ * OUT_;         // BN*H floats (64 KB)
  float* stgt = ssrc + (size_t)BN_ * H_;         // BN*H floats (64 KB)

  wh_kernel<<<BN_ / 16, 256, 0, stream>>>(h, W, Wh);
  score_kernel<<<(BN_ * H_) / 256, 256, 0, stream>>>(Wh, a_s, a_t, ssrc, stgt);
  gat_kernel<<<BN_ / 16, 128, 0, stream>>>(h, ef, Wh, ssrc, stgt, We, gm, bt,
                                           out);
}